// MultiHeadSelfAttention_54039278518668
// MI455X (gfx1250) — compile-verified
//
#include <hip/hip_runtime.h>
#include <hip/hip_bf16.h>

// MI455X / gfx1250, wave32. fp32 WMMA (16x16x4): problem is bandwidth-bound
// (~0.66 GB HBM vs 137 GFLOP), so fp32 matrix rate is free and matches the
// fp32 reference numerics. GEMMs are LDS-tiled (128x128/block, double-buffered
// async copies, ASYNCcnt) to cut L2 traffic ~3x vs streaming fragments.

typedef __attribute__((ext_vector_type(2))) float v2f;
typedef __attribute__((ext_vector_type(8))) float v8f;

static constexpr int Bb = 8, Nn = 2048, Ee = 1024, Hh = 16, Dd = 64;

__device__ __forceinline__ v8f wmma4(v2f a, v2f b, v8f c) {
  // (neg_a, A, neg_b, B, c_mod, C, reuse_a, reuse_b)
  return __builtin_amdgcn_wmma_f32_16x16x4_f32(false, a, false, b, (short)0, c,
                                               false, false);
}

// Async DMA: 16 bytes global -> LDS, tracked by ASYNCcnt (ISA 15.18.3 op 98).
__device__ __forceinline__ void async_b128(const float* g, const float* lds) {
  unsigned l = (unsigned)(uintptr_t)lds;  // low 32 bits of generic ptr = LDS addr
  asm volatile("global_load_async_to_lds_b128 %0, %1, off"
               :: "v"(l), "v"(g) : "memory");
}

// ---------------------------------------------------------------------------
// C[M,N] = A[M,K] @ W[N,K]^T + bias      (torch Linear: x @ W.T + b)
// Block (256 thr = 8 waves): 128x128 tile. Waves 4(M) x 2(N), each 32x64.
// K staged in double-buffered LDS chunks of 16 via async-to-LDS B128 copies.
// LDS row stride 20 floats: 16B-aligned B128 stores, conflict-free v2f reads.
// ---------------------------------------------------------------------------
__global__ __launch_bounds__(256) void gemm_xwt_bias_lds(
    const float* __restrict__ A, const float* __restrict__ W,
    const float* __restrict__ bias, float* __restrict__ C,
    int M, int N, int K) {
  constexpr int KB = 16, PAD = 20;
  __shared__ float sA[2][128][PAD];
  __shared__ float sB[2][128][PAD];

  const int tid  = threadIdx.x;
  const int lane = tid & 31;
  const int wave = tid >> 5;      // 0..7
  const int wm   = wave >> 1;     // 0..3 : 32-row slice
  const int wn   = wave & 1;      // 0..1 : 64-col slice
  const int lr = lane & 15, hi = lane >> 4, kkb = hi << 1;

  const long row0 = (long)blockIdx.y * 128;
  const long col0 = (long)blockIdx.x * 128;

  // Loader mapping: 512 vec4 per 128x16 chunk; thread t moves rows r0, r0+64.
  const int r0 = tid >> 2;            // 0..63
  const int c4 = (tid & 3) << 2;      // 0,4,8,12
  const float* gA0 = A + (row0 + r0) * (long)K + c4;
  const float* gA1 = gA0 + 64l * K;
  const float* gB0 = W + (col0 + r0) * (long)K + c4;
  const float* gB1 = gB0 + 64l * K;

  auto issue = [&](int kc, int buf) {
    const long ko = (long)kc * KB;
    async_b128(gA0 + ko, &sA[buf][r0][c4]);
    async_b128(gA1 + ko, &sA[buf][64 + r0][c4]);
    async_b128(gB0 + ko, &sB[buf][r0][c4]);
    async_b128(gB1 + ko, &sB[buf][64 + r0][c4]);
  };

  v8f acc[2][4] = {};
  const int NC = K / KB;
  issue(0, 0);

  for (int kc = 0; kc < NC; ++kc) {
    const int buf = kc & 1;
    if (kc + 1 < NC) {
      issue(kc + 1, buf ^ 1);
      // 4 just-issued outstanding; <=4 means chunk kc's 4 transfers landed
      asm volatile("s_wait_asynccnt 0x4" ::: "memory");
    } else {
      asm volatile("s_wait_asynccnt 0x0" ::: "memory");
    }
    __syncthreads();
#pragma unroll
    for (int kk = 0; kk < KB; kk += 4) {
      v2f a[2], b[4];
#pragma unroll
      for (int mi = 0; mi < 2; ++mi)
        a[mi] = *(const v2f*)&sA[buf][wm * 32 + mi * 16 + lr][kk + kkb];
#pragma unroll
      for (int s = 0; s < 4; ++s)
        b[s] = *(const v2f*)&sB[buf][wn * 64 + s * 16 + lr][kk + kkb];
#pragma unroll
      for (int mi = 0; mi < 2; ++mi)
#pragma unroll
        for (int s = 0; s < 4; ++s) acc[mi][s] = wmma4(a[mi], b[s], acc[mi][s]);
    }
    __syncthreads();  // all waves done with buf before it is refilled
  }

  // Epilogue: C/D layout -> row = r + 8*hi (per acc VGPR r), col = lr.
#pragma unroll
  for (int s = 0; s < 4; ++s) {
    const long col = col0 + wn * 64 + s * 16 + lr;
    const float bv = bias[col];
#pragma unroll
    for (int mi = 0; mi < 2; ++mi) {
      const long rbase = row0 + wm * 32 + mi * 16 + 8 * hi;
#pragma unroll
      for (int r = 0; r < 8; ++r)
        C[(rbase + r) * (long)N + col] = acc[mi][s][r] + bv;
    }
  }
}

// ---------------------------------------------------------------------------
// Per-(b,t) head-mixing attention (replicates the reference einsum exactly).
// qkv row (3072 floats) = Q(16x64) | K(16x64) | V(16x64).
//   energy = Q @ K^T (16x16), softmax over columns (head axis, /sqrt(E)),
//   out(16x64) = attn @ V, stored into [B,H,N,D]-ordered buffer so the final
//   GEMM reads the same raw-reshape the reference performs.
// One wave per token; 8 waves / 256-thread block.
// ---------------------------------------------------------------------------
__global__ __launch_bounds__(256) void attn_headmix(
    const float* __restrict__ qkv, float* __restrict__ outBHND) {
  __shared__ float sA[8 * 256];  // 16x16 attention tile per wave
  const int lane = threadIdx.x & 31;
  const int wib  = threadIdx.x >> 5;
  const int wid  = blockIdx.x * 8 + wib;
  const int lr = lane & 15, hi = lane >> 4, kkb = hi << 1;
  const int b = wid >> 11;      // / 2048
  const int t = wid & 2047;

  const float* base = qkv + (size_t)wid * (3 * Ee);
  const float* Q  = base;
  const float* Kp = base + Ee;
  const float* V  = base + 2 * Ee;

  // energy = Q(16x64) @ K(16x64)^T : B[k,j] = K[j,k] -> contiguous per lane.
  v8f e = {};
#pragma unroll
  for (int k0 = 0; k0 < 64; k0 += 4) {
    v2f a  = *(const v2f*)(Q  + lr * 64 + k0 + kkb);
    v2f bq = *(const v2f*)(Kp + lr * 64 + k0 + kkb);
    e = wmma4(a, bq, e);
  }

  // Softmax over the 16 columns of each row (cols live in one 16-lane half).
  float* sAttnW = sA + wib * 256;
  const float scale = 0.03125f;  // 1/sqrt(1024)
#pragma unroll
  for (int r = 0; r < 8; ++r) {
    float x = e[r] * scale;
    float mx = x;
#pragma unroll
    for (int m = 1; m <= 8; m <<= 1) mx = fmaxf(mx, __shfl_xor(mx, m, 32));
    float p = __expf(x - mx);
    float den = p;
#pragma unroll
    for (int m = 1; m <= 8; m <<= 1) den += __shfl_xor(den, m, 32);
    sAttnW[(r + 8 * hi) * 16 + lr] = p / den;  // attn[i=r+8*hi][l=lr]
  }
  __syncthreads();  // C-layout -> A-layout reshape goes through LDS

  // out(16x64) = attn(16x16) @ V(16x64)
  v8f o[4] = {};
#pragma unroll
  for (int k0 = 0; k0 < 16; k0 += 4) {
    v2f a = *(const v2f*)(sAttnW + lr * 16 + k0 + kkb);  // A[i=lr, l=k]
#pragma unroll
    for (int s = 0; s < 4; ++s) {
      v2f bv;
      bv.x = V[(k0 + kkb)     * 64 + s * 16 + lr];  // B[k,d] = V[k,d]
      bv.y = V[(k0 + kkb + 1) * 64 + s * 16 + lr];
      o[s] = wmma4(a, bv, o[s]);
    }
  }

  // Store out[b,i,t,d] at flat ((b*16+i)*2048 + t)*64 + d  ([B,H,N,D] order).
  float* ob = outBHND + (size_t)b * (Hh * Nn * Dd) + (size_t)t * Dd;
#pragma unroll
  for (int s = 0; s < 4; ++s)
#pragma unroll
    for (int r = 0; r < 8; ++r) {
      const int i = r + 8 * hi;
      const int d = s * 16 + lr;
      ob[(size_t)i * (Nn * Dd) + d] = o[s][r];
    }
}

// ---------------------------------------------------------------------------
extern "C" void kernel_launch(void* const* d_in, const int* in_sizes, int n_in,
                              void* d_out, int out_size, void* d_ws,
                              size_t ws_size, hipStream_t stream) {
  const float* x    = (const float*)d_in[0];  // [8,2048,1024]
  const float* Wqkv = (const float*)d_in[1];  // [3072,1024]
  const float* bqkv = (const float*)d_in[2];  // [3072]
  const float* Wo   = (const float*)d_in[3];  // [1024,1024]
  const float* bo   = (const float*)d_in[4];  // [1024]
  float* out = (float*)d_out;                 // [8,2048,1024]

  float* qkv     = (float*)d_ws;                       // 192 MB
  float* attnout = qkv + (size_t)Bb * Nn * 3 * Ee;     // +64 MB

  const int M = Bb * Nn;  // 16384

  // GEMM1: qkv = x @ Wqkv^T + bqkv   [16384 x 3072], K=1024
  gemm_xwt_bias_lds<<<dim3(3 * Ee / 128, M / 128), 256, 0, stream>>>(
      x, Wqkv, bqkv, qkv, M, 3 * Ee, Ee);

  // Attention: one wave per (b,t): 16384 waves -> 2048 blocks
  attn_headmix<<<2048, 256, 0, stream>>>(qkv, attnout);

  // GEMM2: out = attnout(as [16384,1024]) @ Wo^T + bo
  gemm_xwt_bias_lds<<<dim3(Ee / 128, M / 128), 256, 0, stream>>>(
      attnout, Wo, bo, out, M, Ee, Ee);
}